// BSpline_22213570855055
// MI455X (gfx1250) — compile-verified
//
#include <hip/hip_runtime.h>

typedef __attribute__((ext_vector_type(2))) float v2f;
typedef __attribute__((ext_vector_type(8))) float v8f;

#define DEG 3
#define NGROUPS 4

// Interleaved transpose: cTi[p*2*dim + 2*d + e] = c[d*nk + 2*p + e]
// A lane's (k,k+1) B-operand pair becomes one b64, and consecutive lanes
// (d, d+1) are adjacent -> fully coalesced 256B/wave B loads.
__global__ void bspline_transpose_kernel(const float* __restrict__ c,
                                         float* __restrict__ cTi,
                                         int dim, int nk) {
    int j = blockIdx.x * blockDim.x + threadIdx.x;
    int total = dim * nk;
    if (j < total) {
        int e = j & 1;
        int d = (j >> 1) % dim;
        int p = j / (2 * dim);
        cTi[j] = c[d * nk + 2 * p + e];
    }
}

__device__ __forceinline__ float pick4(float w0, float w1, float w2, float w3, int idx) {
    float r = 0.0f;
    r = (idx == 0) ? w0 : r;
    r = (idx == 1) ? w1 : r;
    r = (idx == 2) ? w2 : r;
    r = (idx == 3) ? w3 : r;
    return r;
}

// One wave32 per block: blockIdx.x = 16-sample tile (scalar), blockIdx.y = dim group.
// CDIM != 0: compile-time dim -> load/store offsets become instruction immediates.
template <int CDIM>
__global__ void bspline_wmma_kernel(const float* __restrict__ t,
                                    const float* __restrict__ c,    // [dim, nk]
                                    const float* __restrict__ cTi,  // interleaved c^T
                                    const int* __restrict__ deltaPtr,
                                    float* __restrict__ outT,       // [S]
                                    float* __restrict__ outV,       // [S, dim]
                                    int S, int nk, int dimArg, int useCT,
                                    int chunksPerGroup) {
    const int dim = CDIM ? CDIM : dimArg;

    const int lane  = threadIdx.x;       // blockDim.x == 32
    const int tile  = blockIdx.x;        // scalar (SGPR)
    const int group = blockIdx.y;        // scalar (SGPR)

    const int dimChunks = dim >> 4;
    const int dc0 = group * chunksPerGroup;
    if (dc0 >= dimChunks) return;        // scalar branch
    const int dc1 = min(dc0 + chunksPerGroup, dimChunks);

    const int sid  = lane & 15;          // sample row within tile == dim col within chunk
    const int half = lane >> 4;          // k-half selector for A/B operand layout
    const int s = tile * 16 + sid;

    const float delta = (float)deltaPtr[0];
    const float x = (float)s * delta;

    // span search: first j with t[j] > x among j in [0, nk-DEG); m = DEG + j
    const int cand = nk - DEG;
    int lo = 0, cnt = cand;
    while (cnt > 0) {
        int step = cnt >> 1;
        int mid = lo + step;
        if (t[mid] <= x) { lo = mid + 1; cnt -= step + 1; }
        else             { cnt = step; }
    }
    const bool oor = (s >= S) || (lo >= cand);  // past padded_t[nk] -> zero row (matches ref)
    const int m = DEG + (lo < cand - 1 ? lo : cand - 1);

    // local de Boor weights N[0..3] = B_{m-3..m}^3(x); where(denom==0,0) convention
    float N0, N1, N2, N3;
    {
        float N[4] = {1.0f, 0.0f, 0.0f, 0.0f};
        float left[4], right[4];
#pragma unroll
        for (int j = 1; j <= DEG; ++j) {
            int iL = m + 1 - j, iR = m + j;
            left[j]  = x - ((iL < DEG + 1) ? 0.0f : t[iL - (DEG + 1)]);
            right[j] = ((iR < DEG + 1) ? 0.0f : t[iR - (DEG + 1)]) - x;
            float saved = 0.0f;
#pragma unroll
            for (int r = 0; r < j; ++r) {
                float den = right[r + 1] + left[j - r];
                float temp = (den != 0.0f) ? (N[r] / den) : 0.0f;
                N[r] = saved + right[r + 1] * temp;
                saved = left[j - r] * temp;
            }
            N[j] = saved;
        }
        if (oor) N[0] = N[1] = N[2] = N[3] = 0.0f;
        N0 = N[0]; N1 = N[1]; N2 = N[2]; N3 = N[3];
    }

    if (group == 0 && half == 0 && s < S) outT[s] = x;

    // wave min/max span over the 16 samples -> force scalar (all lanes converged)
    int mn = m, mx = m;
#pragma unroll
    for (int off = 16; off >= 1; off >>= 1) {
        mn = min(mn, __shfl_xor(mn, off, 32));
        mx = max(mx, __shfl_xor(mx, off, 32));
    }
    mn = __builtin_amdgcn_readfirstlane(mn);
    mx = __builtin_amdgcn_readfirstlane(mx);

    // even base, window always inside [0, nk): no bounds checks in hot loop
    int base = mn - DEG;
    const int bmax = nk - 16;
    base = (base < bmax ? base : bmax) & ~1;
    const bool wide = (mx - base) > 15;            // scalar
    const bool fullTile = (tile * 16 + 16) <= S;   // scalar

    if (!wide) {
        // A operand for 4 K-chunks (16x4 f32 layout: v0 = k0|k2, v1 = k1|k3)
        const int offS = m - DEG - base;           // in [0, 13]
        v2f a[4];
#pragma unroll
        for (int kc = 0; kc < 4; ++kc) {
            int col0 = 4 * kc + 2 * half;
            a[kc].x = pick4(N0, N1, N2, N3, col0 - offS);
            a[kc].y = pick4(N0, N1, N2, N3, col0 + 1 - offS);
        }

        if (useCT) {
            // B from interleaved c^T: per-K-chunk offsets are compile-time when CDIM!=0
            const float* bp = cTi + base * dim + half * 2 * dim + 2 * sid + dc0 * 32;
            float* op = outV + (tile * 16 + 8 * half) * dim + sid + (dc0 << 4);
            if (fullTile) {
#pragma unroll 2
                for (int dc = dc0; dc < dc1; ++dc) {
                    v8f acc = {0.f, 0.f, 0.f, 0.f, 0.f, 0.f, 0.f, 0.f};
#pragma unroll
                    for (int kc = 0; kc < 4; ++kc) {
                        v2f B = *(const v2f*)(bp + kc * (4 * dim));
                        acc = __builtin_amdgcn_wmma_f32_16x16x4_f32(
                                  false, a[kc], false, B, (short)0, acc, false, false);
                    }
                    bp += 32;
#pragma unroll
                    for (int r = 0; r < 8; ++r)
                        op[r * dim] = acc[r];
                    op += 16;
                }
            } else {
#pragma unroll 1
                for (int dc = dc0; dc < dc1; ++dc) {
                    v8f acc = {0.f, 0.f, 0.f, 0.f, 0.f, 0.f, 0.f, 0.f};
#pragma unroll
                    for (int kc = 0; kc < 4; ++kc) {
                        v2f B = *(const v2f*)(bp + kc * (4 * dim));
                        acc = __builtin_amdgcn_wmma_f32_16x16x4_f32(
                                  false, a[kc], false, B, (short)0, acc, false, false);
                    }
                    bp += 32;
#pragma unroll
                    for (int r = 0; r < 8; ++r) {
                        int srow = tile * 16 + r + 8 * half;
                        if (srow < S) op[r * dim] = acc[r];
                    }
                    op += 16;
                }
            }
        } else {
            // B directly from c: per-K-chunk offsets are constants (kc*4)
            const float* bp = c + sid * nk + base + 2 * half + dc0 * 16 * nk;
            float* op = outV + (tile * 16 + 8 * half) * dim + sid + (dc0 << 4);
#pragma unroll 1
            for (int dc = dc0; dc < dc1; ++dc) {
                v8f acc = {0.f, 0.f, 0.f, 0.f, 0.f, 0.f, 0.f, 0.f};
#pragma unroll
                for (int kc = 0; kc < 4; ++kc) {
                    v2f B = *(const v2f*)(bp + kc * 4);
                    acc = __builtin_amdgcn_wmma_f32_16x16x4_f32(
                              false, a[kc], false, B, (short)0, acc, false, false);
                }
                bp += 16 * nk;
                if (fullTile) {
#pragma unroll
                    for (int r = 0; r < 8; ++r)
                        op[r * dim] = acc[r];
                } else {
#pragma unroll
                    for (int r = 0; r < 8; ++r) {
                        int srow = tile * 16 + r + 8 * half;
                        if (srow < S) op[r * dim] = acc[r];
                    }
                }
                op += 16;
            }
        }
    } else {
        // rare wide-window fallback: lane evaluates its OWN sample for 8 dims per chunk;
        // support columns m-3..m are contiguous and always in [0, nk)
        const bool live = (s < S);
        const int crow = m - DEG;
#pragma unroll 1
        for (int dc = dc0; dc < dc1; ++dc) {
            int dbase = (dc << 4) + half * 8;
#pragma unroll
            for (int j2 = 0; j2 < 8; ++j2) {
                int d = dbase + j2;
                int ci = d * nk + crow;
                float v = N0 * c[ci] + N1 * c[ci + 1] + N2 * c[ci + 2] + N3 * c[ci + 3];
                if (live) outV[s * dim + d] = v;
            }
        }
    }
}

extern "C" void kernel_launch(void* const* d_in, const int* in_sizes, int n_in,
                              void* d_out, int out_size, void* d_ws, size_t ws_size,
                              hipStream_t stream) {
    const float* t     = (const float*)d_in[0];
    const float* c     = (const float*)d_in[1];
    const int*   delta = (const int*)d_in[2];

    const int nk  = in_sizes[0];            // 2048
    const int dim = in_sizes[1] / nk;       // 512
    const int S   = out_size / (dim + 1);   // 32768

    float* outT = (float*)d_out;
    float* outV = outT + S;

    const size_t ctBytes = (size_t)nk * (size_t)dim * sizeof(float);
    const int useCT = (ws_size >= ctBytes && (nk % 2 == 0)) ? 1 : 0;
    float* cTi = (float*)d_ws;

    if (useCT) {
        int total = nk * dim;
        bspline_transpose_kernel<<<(total + 255) / 256, 256, 0, stream>>>(c, cTi, dim, nk);
    }

    const int sTiles = (S + 15) / 16;                 // one wave32 tile per blockIdx.x
    const int dimChunks = dim / 16;
    const int chunksPerGroup = (dimChunks + NGROUPS - 1) / NGROUPS;

    dim3 grid(sTiles, NGROUPS, 1);
    if (dim == 512) {
        bspline_wmma_kernel<512><<<grid, 32, 0, stream>>>(
            t, c, cTi, delta, outT, outV, S, nk, dim, useCT, chunksPerGroup);
    } else {
        bspline_wmma_kernel<0><<<grid, 32, 0, stream>>>(
            t, c, cTi, delta, outT, outV, S, nk, dim, useCT, chunksPerGroup);
    }
}